// Numeral_Machine_35055523070757
// MI455X (gfx1250) — compile-verified
//
#include <hip/hip_runtime.h>
#include <math.h>

typedef __attribute__((ext_vector_type(2))) float v2f;
typedef __attribute__((ext_vector_type(8))) float v8f;

#define ORDER   20
#define LMOD    5.0f
#define VOCAB   10
#define HID     100
#define EPSF    1e-10f
#define TWO_PI_OVER_L 1.2566370614359172f   /* 2*pi/5   */
#define L_OVER_PI     1.5915494309189535f   /* 5/pi     */

__device__ __forceinline__ float sigmoidf_fast(float z) {
    return 1.0f / (1.0f + __expf(-z));
}

// One wave (32 lanes) processes 32 consecutive rows.
// Lane l owns row (rowbase + l) for the scalar NALU chain.
// WMMA tiles: tile0 = rows [rowbase, rowbase+16), tile1 = rows [rowbase+16, rowbase+32).
__global__ __launch_bounds__(256) void numeral_machine_kernel(
    const float* __restrict__ x_in,   // [B,1]
    const float* __restrict__ gum,    // [8,B,10]
    const float* __restrict__ subW,   // (2,3)
    const float* __restrict__ subM,   // (2,3)
    const float* __restrict__ divW,   // (1,2)
    const float* __restrict__ divM,   // (1,2)
    const float* __restrict__ divG,   // (1,2)
    const float* __restrict__ w1,     // (100,1)
    const float* __restrict__ b1,     // (100,)
    const float* __restrict__ w2,     // (10,100)
    const float* __restrict__ b2,     // (10,)
    const float* __restrict__ wr,     // (1,11)
    const float* __restrict__ br,     // (1,)
    float* __restrict__ out,          // [B,1]
    int B)
{
    __shared__ __align__(16) float s_w1[HID];
    __shared__ __align__(16) float s_b1[HID];
    __shared__ __align__(16) float s_w2[16 * HID];   // zero-padded rows 10..15
    __shared__ float s_wr[VOCAB];

    const int tid = threadIdx.x;
    for (int idx = tid; idx < HID; idx += 256) {
        s_w1[idx] = w1[idx];
        s_b1[idx] = b1[idx];
    }
    for (int idx = tid; idx < 16 * HID; idx += 256) {
        int r = idx / HID, c = idx - r * HID;
        s_w2[idx] = (r < VOCAB) ? w2[r * HID + c] : 0.0f;
    }
    if (tid < VOCAB) s_wr[tid] = wr[tid];
    __syncthreads();

    const int lane = tid & 31;
    const int n    = lane & 15;     // WMMA column (B, D) / A row id
    const int half = lane >> 4;     // 0: K={0,1}, 1: K={2,3} within a chunk
    const int wave = tid >> 5;
    const int rowbase = blockIdx.x * 256 + wave * 32;
    const int myrow   = rowbase + lane;

    // Effective NAC/NALU weights (constant; computed once per lane).
    // Note: sub weights are (2,3) but the concat input is [x, mod] (2 cols);
    // we use the first two columns of each row (self-consistent forward).
    const float Ws00 = tanhf(subW[0]) * sigmoidf_fast(subM[0]);
    const float Ws01 = tanhf(subW[1]) * sigmoidf_fast(subM[1]);
    const float Ws10 = tanhf(subW[3]) * sigmoidf_fast(subM[3]);
    const float Ws11 = tanhf(subW[4]) * sigmoidf_fast(subM[4]);
    const float Wd0  = tanhf(divW[0]) * sigmoidf_fast(divM[0]);
    const float Wd1  = tanhf(divW[1]) * sigmoidf_fast(divM[1]);
    const float G0   = divG[0], G1 = divG[1];
    const float wr10 = wr[VOCAB];
    const float brv  = br[0];
    const float b2v  = (n < VOCAB) ? b2[n] : 0.0f;

    float xv = x_in[myrow];

    // Receiver accumulators, kept in WMMA D-layout:
    // hacc{t}[v] corresponds to row = rowbase + 16*t + v + 8*half
    float hacc0[8], hacc1[8];
    #pragma unroll
    for (int v = 0; v < 8; ++v) { hacc0[v] = 0.0f; hacc1[v] = 0.0f; }

    float pw = 1.0f;   // wr10^i

    for (int i = 0; i < 8; ++i) {
        // ---- smooth_mod(x + 0.5): 20 sines ----
        const float w = (xv + 0.5f) * TWO_PI_OVER_L;
        float term = 0.0f;
        #pragma unroll
        for (int nn = 1; nn <= ORDER; ++nn)
            term += __sinf(w * (float)nn) * (1.0f / (float)nn);
        const float mod = 0.5f * LMOD - L_OVER_PI * term;

        // Broadcast mod of the row each lane services in each 16-row tile.
        const float mod0 = __shfl(mod, n);        // row n      (tile 0)
        const float mod1 = __shfl(mod, n + 16);   // row n + 16 (tile 1)

        // ---- vs = relu(mod*w1 + b1) @ w2_pad  via V_WMMA_F32_16X16X4_F32 ----
        v8f acc0 = {}; v8f acc1 = {};
        #pragma unroll
        for (int c = 0; c < HID / 4; ++c) {
            const int k = 4 * c + 2 * half;       // even -> b64-aligned LDS reads
            const v2f w1p = *(const v2f*)&s_w1[k];
            const v2f b1p = *(const v2f*)&s_b1[k];
            const v2f bfr = *(const v2f*)&s_w2[n * HID + k];  // B: w2_pad[n][k..k+1]

            v2f a0, a1;
            a0.x = fmaxf(fmaf(mod0, w1p.x, b1p.x), 0.0f);
            a0.y = fmaxf(fmaf(mod0, w1p.y, b1p.y), 0.0f);
            a1.x = fmaxf(fmaf(mod1, w1p.x, b1p.x), 0.0f);
            a1.y = fmaxf(fmaf(mod1, w1p.y, b1p.y), 0.0f);

            acc0 = __builtin_amdgcn_wmma_f32_16x16x4_f32(
                false, a0, false, bfr, (short)0, acc0, false, false);
            acc1 = __builtin_amdgcn_wmma_f32_16x16x4_f32(
                false, a1, false, bfr, (short)0, acc1, false, false);
        }

        // ---- gumbel + argmax (per row), fold into receiver accumulator ----
        // D layout: acc[v] @ lane l = vs[row = 16*t + v + 8*half][col = n]
        #pragma unroll
        for (int v = 0; v < 8; ++v) {
            // tile 0
            {
                const int row = rowbase + v + 8 * half;
                float u = (n < VOCAB) ? gum[((long)i * B + row) * VOCAB + n] : 0.5f;
                float g = -__logf(-__logf(u + EPSF) + EPSF);
                float logit = (n < VOCAB) ? (acc0[v] + b2v + g) : -__builtin_inff();
                int   bi = n;
                #pragma unroll
                for (int m = 1; m < 16; m <<= 1) {
                    float ov = __shfl_xor(logit, m);
                    int   oi = __shfl_xor(bi, m);
                    if (ov > logit || (ov == logit && oi < bi)) { logit = ov; bi = oi; }
                }
                hacc0[v] += (s_wr[bi] + brv) * pw;
            }
            // tile 1
            {
                const int row = rowbase + 16 + v + 8 * half;
                float u = (n < VOCAB) ? gum[((long)i * B + row) * VOCAB + n] : 0.5f;
                float g = -__logf(-__logf(u + EPSF) + EPSF);
                float logit = (n < VOCAB) ? (acc1[v] + b2v + g) : -__builtin_inff();
                int   bi = n;
                #pragma unroll
                for (int m = 1; m < 16; m <<= 1) {
                    float ov = __shfl_xor(logit, m);
                    int   oi = __shfl_xor(bi, m);
                    if (ov > logit || (ov == logit && oi < bi)) { logit = ov; bi = oi; }
                }
                hacc1[v] += (s_wr[bi] + brv) * pw;
            }
        }

        // ---- per-lane NAC -> NALU chain to update x for next iteration ----
        const float h0 = Ws00 * xv + Ws01 * mod;
        const float h1 = Ws10 * xv + Ws11 * mod;
        const float a  = Wd0 * h0 + Wd1 * h1;
        const float gg = sigmoidf_fast(G0 * h0 + G1 * h1);
        const float mm = __expf(Wd0 * __logf(fabsf(h0) + EPSF) +
                                Wd1 * __logf(fabsf(h1) + EPSF));
        xv = gg * a + (1.0f - gg) * mm;

        pw *= wr10;
    }

    // ---- write out: one writer lane per row ----
    #pragma unroll
    for (int v = 0; v < 8; ++v) {
        if (n == v) {
            out[rowbase +      v + 8 * half] = hacc0[v];
            out[rowbase + 16 + v + 8 * half] = hacc1[v];
        }
    }
}

extern "C" void kernel_launch(void* const* d_in, const int* in_sizes, int n_in,
                              void* d_out, int out_size, void* d_ws, size_t ws_size,
                              hipStream_t stream) {
    const float* x    = (const float*)d_in[0];
    const float* gum  = (const float*)d_in[1];
    const float* subW = (const float*)d_in[2];
    const float* subM = (const float*)d_in[3];
    const float* divW = (const float*)d_in[4];
    const float* divM = (const float*)d_in[5];
    const float* divG = (const float*)d_in[6];
    const float* w1   = (const float*)d_in[7];
    const float* b1   = (const float*)d_in[8];
    const float* w2   = (const float*)d_in[9];
    const float* b2   = (const float*)d_in[10];
    const float* wr   = (const float*)d_in[11];
    const float* br   = (const float*)d_in[12];
    float* out = (float*)d_out;

    const int B = in_sizes[0];            // 131072 (divisible by 256)
    const int blocks = (B + 255) / 256;

    numeral_machine_kernel<<<blocks, 256, 0, stream>>>(
        x, gum, subW, subM, divW, divM, divG, w1, b1, w2, b2, wr, br, out, B);
}